// _FallbackGConvGRU_49838800503662
// MI455X (gfx1250) — compile-verified
//
#include <hip/hip_runtime.h>
#include <hip/hip_bf16.h>

typedef __attribute__((ext_vector_type(2))) float v2f;
typedef __attribute__((ext_vector_type(8))) float v8f;

#define NN 50000
#define EE 800000
#define CIN 64
#define COUT 64
#define GC 128      // gate input channels (x | h)
#define KTOT 256    // concat K dim for [Tx0 | Tx1]
#define PKW (4 * 64 * 32 * 2)   // packed weight size = 16384 floats

// ---------------- init / edge kernels ----------------

__global__ void k_zero(float* p, int n) {
    int i = blockIdx.x * blockDim.x + threadIdx.x;
    if (i < n) p[i] = 0.0f;
}

// Build G = [x | h | 0] as [N, 256]
__global__ void k_build_gate(const float* __restrict__ x,
                             const float* __restrict__ h,
                             float* __restrict__ G) {
    unsigned tid = blockIdx.x * blockDim.x + threadIdx.x;
    if (tid >= (unsigned)NN * KTOT) return;
    int i = tid >> 8;        // node
    int c = tid & 255;       // column
    float v;
    if (c < CIN)        v = x[i * CIN + c];
    else if (c < GC)    v = h[i * COUT + (c - CIN)];
    else                v = 0.0f;
    G[(unsigned)i * KTOT + c] = v;
}

// Rebuild candidate input in-place: cols 64..127 = r*h, cols 128..255 = 0 (x cols kept)
__global__ void k_build_cand(const float* __restrict__ r,
                             const float* __restrict__ h,
                             float* __restrict__ G) {
    unsigned tid = blockIdx.x * blockDim.x + threadIdx.x;
    if (tid >= (unsigned)NN * KTOT) return;
    int i = tid >> 8;
    int c = tid & 255;
    if (c < CIN) return;
    if (c < GC) {
        int cc = c - CIN;
        G[(unsigned)i * KTOT + c] = r[i * COUT + cc] * h[i * COUT + cc];
    } else {
        G[(unsigned)i * KTOT + c] = 0.0f;
    }
}

__global__ void k_degree(const int* __restrict__ ei,
                         const float* __restrict__ w,
                         float* __restrict__ deg) {
    int e = blockIdx.x * blockDim.x + threadIdx.x;
    if (e >= EE) return;
    atomicAdd(&deg[ei[e]], w[e]);
}

__global__ void k_dinv(const float* __restrict__ deg, float* __restrict__ dinv) {
    int i = blockIdx.x * blockDim.x + threadIdx.x;
    if (i >= NN) return;
    float d = deg[i];
    dinv[i] = d > 0.0f ? rsqrtf(d) : 0.0f;
}

__global__ void k_lw(const int* __restrict__ ei,
                     const float* __restrict__ w,
                     const float* __restrict__ dinv,
                     float* __restrict__ lw) {
    int e = blockIdx.x * blockDim.x + threadIdx.x;
    if (e >= EE) return;
    int ro = ei[e];
    int co = ei[EE + e];
    lw[e] = -w[e] * dinv[ro] * dinv[co];
}

// prop: G[row, 128+c] += lw[e] * G[col, c]   for c in [0,128)
// One 32-thread workgroup per edge -> ei/lw are block-uniform (scalar loads),
// each lane gathers a float4 (coalesced b128) and issues 4 fp32 atomics.
__global__ void __launch_bounds__(32)
k_scatter(const int* __restrict__ ei,
          const float* __restrict__ lw,
          float* __restrict__ G) {
    int e  = blockIdx.x;
    int ro = ei[e];
    int co = ei[EE + e];
    float lwv = lw[e];
    int c = threadIdx.x * 4;
    float4 g = *reinterpret_cast<const float4*>(G + (unsigned)co * KTOT + c);
    float* dst = G + (unsigned)ro * KTOT + GC + c;
    atomicAdd(dst + 0, lwv * g.x);
    atomicAdd(dst + 1, lwv * g.y);
    atomicAdd(dst + 2, lwv * g.z);
    atomicAdd(dst + 3, lwv * g.w);
}

// ---------------- weight repack (fragment order) ----------------
// P[((ct*64 + k)*32 + lane)*2 + {0,1}] = W[kb][cb+n], W[kb+1][cb+n]
// with kb = 4k + 2*(lane>>4), n = lane&15, cb = ct*16. W viewed as [256,64].
__global__ void k_pack_w(const float* __restrict__ W, float* __restrict__ P) {
    int tid = blockIdx.x * blockDim.x + threadIdx.x;
    if (tid >= 4 * 64 * 32) return;
    int lane = tid & 31;
    int k    = (tid >> 5) & 63;
    int ct   = tid >> 11;
    int hi = lane >> 4;
    int n  = lane & 15;
    int kb = 4 * k + 2 * hi;
    int cb = ct * 16;
    P[tid * 2 + 0] = W[kb * COUT + cb + n];
    P[tid * 2 + 1] = W[(kb + 1) * COUT + cb + n];
}

// ---------------- WMMA GEMM kernels ----------------

__device__ __forceinline__ float fast_sigmoid(float x) {
    // v_exp_f32 + v_rcp_f32 (avoid IEEE div_scale/div_fmas chain)
    return __builtin_amdgcn_rcpf(1.0f + __expf(-x));
}
__device__ __forceinline__ float fast_tanh(float x) {
    float t = __expf(2.0f * x);
    return 1.0f - 2.0f * __builtin_amdgcn_rcpf(t + 1.0f);
}

// 16-row tile per block; 8 waves: waves 0..3 -> z (Pz), waves 4..7 -> r (Pr).
__global__ void __launch_bounds__(256)
k_gemm_gates(const float* __restrict__ G,
             const float* __restrict__ Pz, const float* __restrict__ bz,
             const float* __restrict__ Pr, const float* __restrict__ br,
             float* __restrict__ zbuf, float* __restrict__ rbuf) {
    int row0 = blockIdx.x * 16;
    int wave = threadIdx.x >> 5;
    int lane = threadIdx.x & 31;
    int ct   = wave & 3;          // 16-col tile within the 64 output cols
    bool isR = wave >= 4;
    const float* P = isR ? Pr : Pz;
    const float* B = isR ? br : bz;
    float* out = isR ? rbuf : zbuf;

    int cb  = ct * 16;
    int n   = lane & 15;
    int hi  = lane >> 4;          // 0: K pair (k0,k0+1), 1: (k0+2,k0+3)
    int m16 = lane & 15;

    v8f acc = {0.f, 0.f, 0.f, 0.f, 0.f, 0.f, 0.f, 0.f};

    const float* arow = G + (unsigned)(row0 + m16) * KTOT + 2 * hi;
    const float* brow = P + (unsigned)(ct * 64 * 32 + lane) * 2;
#pragma unroll 4
    for (int k = 0; k < 64; ++k) {
        float2 a2 = *reinterpret_cast<const float2*>(arow + 4 * k);
        float2 b2 = *reinterpret_cast<const float2*>(brow + 64 * k);
        v2f a; a.x = a2.x; a.y = a2.y;
        v2f b; b.x = b2.x; b.y = b2.y;
        acc = __builtin_amdgcn_wmma_f32_16x16x4_f32(
            false, a, false, b, (short)0, acc, false, false);
    }

    float bias = B[cb + n];
#pragma unroll
    for (int v = 0; v < 8; ++v) {
        int m = v + hi * 8;
        out[(unsigned)(row0 + m) * COUT + cb + n] = fast_sigmoid(acc[v] + bias);
    }
}

// 16-row tile per block; 4 waves, one per 16-col tile. Fused GRU update.
__global__ void __launch_bounds__(128)
k_gemm_cand(const float* __restrict__ G,
            const float* __restrict__ Ph, const float* __restrict__ bh,
            const float* __restrict__ zbuf, const float* __restrict__ h,
            float* __restrict__ outp) {
    int row0 = blockIdx.x * 16;
    int wave = threadIdx.x >> 5;
    int lane = threadIdx.x & 31;
    int cb  = wave * 16;
    int n   = lane & 15;
    int hi  = lane >> 4;
    int m16 = lane & 15;

    v8f acc = {0.f, 0.f, 0.f, 0.f, 0.f, 0.f, 0.f, 0.f};

    const float* arow = G + (unsigned)(row0 + m16) * KTOT + 2 * hi;
    const float* brow = Ph + (unsigned)(wave * 64 * 32 + lane) * 2;
#pragma unroll 4
    for (int k = 0; k < 64; ++k) {
        float2 a2 = *reinterpret_cast<const float2*>(arow + 4 * k);
        float2 b2 = *reinterpret_cast<const float2*>(brow + 64 * k);
        v2f a; a.x = a2.x; a.y = a2.y;
        v2f b; b.x = b2.x; b.y = b2.y;
        acc = __builtin_amdgcn_wmma_f32_16x16x4_f32(
            false, a, false, b, (short)0, acc, false, false);
    }

    float bias = bh[cb + n];
#pragma unroll
    for (int v = 0; v < 8; ++v) {
        int m = v + hi * 8;
        unsigned idx = (unsigned)(row0 + m) * COUT + cb + n;
        float ht = fast_tanh(acc[v] + bias);
        float zz = zbuf[idx];
        outp[idx] = (1.0f - zz) * h[idx] + zz * ht;
    }
}

// ---------------- launch ----------------

extern "C" void kernel_launch(void* const* d_in, const int* in_sizes, int n_in,
                              void* d_out, int out_size, void* d_ws, size_t ws_size,
                              hipStream_t stream) {
    const float* x  = (const float*)d_in[0];
    const int*   ei = (const int*)d_in[1];       // [2, E]
    const float* w  = (const float*)d_in[2];
    const float* h  = (const float*)d_in[3];
    const float* Wz = (const float*)d_in[4];     // [2,128,64] contiguous == [256,64]
    const float* bz = (const float*)d_in[5];
    const float* Wr = (const float*)d_in[6];
    const float* br = (const float*)d_in[7];
    const float* Wh = (const float*)d_in[8];
    const float* bh = (const float*)d_in[9];
    float* outp = (float*)d_out;

    float* ws   = (float*)d_ws;
    float* deg  = ws;                       // N
    float* dinv = deg + NN;                 // N
    float* lw   = dinv + NN;                // E
    float* G    = lw + EE;                  // N*256
    float* zbuf = G + (size_t)NN * KTOT;    // N*64
    float* rbuf = zbuf + (size_t)NN * COUT; // N*64
    float* Pz   = rbuf + (size_t)NN * COUT; // 16384
    float* Pr   = Pz + PKW / 2 * 0 + 16384; // 16384
    float* Ph   = Pr + 16384;               // 16384

    const int T = 256;
    // degree + normalized edge weights + packed weights
    k_zero<<<(NN + T - 1) / T, T, 0, stream>>>(deg, NN);
    k_build_gate<<<((unsigned)NN * KTOT + T - 1) / T, T, 0, stream>>>(x, h, G);
    k_pack_w<<<(4 * 64 * 32 + T - 1) / T, T, 0, stream>>>(Wz, Pz);
    k_pack_w<<<(4 * 64 * 32 + T - 1) / T, T, 0, stream>>>(Wr, Pr);
    k_pack_w<<<(4 * 64 * 32 + T - 1) / T, T, 0, stream>>>(Wh, Ph);
    k_degree<<<(EE + T - 1) / T, T, 0, stream>>>(ei, w, deg);
    k_dinv<<<(NN + T - 1) / T, T, 0, stream>>>(deg, dinv);
    k_lw<<<(EE + T - 1) / T, T, 0, stream>>>(ei, w, dinv, lw);

    // Tx1 for gate input
    k_scatter<<<EE, 32, 0, stream>>>(ei, lw, G);

    // z, r gates (WMMA)
    k_gemm_gates<<<NN / 16, 256, 0, stream>>>(G, Pz, bz, Pr, br, zbuf, rbuf);

    // candidate input + its Tx1
    k_build_cand<<<((unsigned)NN * KTOT + T - 1) / T, T, 0, stream>>>(rbuf, h, G);
    k_scatter<<<EE, 32, 0, stream>>>(ei, lw, G);

    // candidate GEMM + fused GRU update (WMMA)
    k_gemm_cand<<<NN / 16, 128, 0, stream>>>(G, Ph, bh, zbuf, h, outp);
}